// VectorQuantizer_1219770712646
// MI455X (gfx1250) — compile-verified
//
#include <hip/hip_runtime.h>

typedef __attribute__((ext_vector_type(2))) float v2f;
typedef __attribute__((ext_vector_type(8))) float v8f;

#define N_VECS   16384
#define K_CODES  8192
#define DIM      256
#define DECAY_F  0.99f
#define OMD_F    0.01f
#define EPS_F    1e-5f
#define COMMIT_F 0.25f

// ---------------- per-code squared norms ----------------
__global__ void vq_enorm(const float* __restrict__ emb, float* __restrict__ enorm) {
    const int k = blockIdx.x;
    const float* row = emb + (size_t)k * DIM;
    float s = 0.f;
#pragma unroll
    for (int i = 0; i < DIM / 32; ++i) {
        float v = row[threadIdx.x + 32 * i];
        s += v * v;
    }
#pragma unroll
    for (int off = 16; off >= 1; off >>= 1)
        s += __shfl_xor(s, off, 32);
    if (threadIdx.x == 0) enorm[k] = s;
}

// ---------------- workspace zeroing ----------------
__global__ void vq_zero(float* __restrict__ p, int n) {
    int i = blockIdx.x * blockDim.x + threadIdx.x;
    if (i < n) p[i] = 0.f;
}

// ---------------- fused GEMM(-2*z@e^T) + argmin via f32 WMMA ----------------
// Block: 256 threads = 8 waves. Block owns 16 rows of z; waves stride over
// the 512 code tiles of 16. A tile (16x256) preloaded once into registers.
// Four independent accumulator chains hide WMMA latency.
__global__ __launch_bounds__(256) void vq_argmin(const float* __restrict__ z,
                                                 const float* __restrict__ emb,
                                                 const float* __restrict__ enorm,
                                                 int* __restrict__ codes_i,
                                                 float* __restrict__ codes_f) {
    __shared__ float s_val[8][16];
    __shared__ int   s_idx[8][16];

    const int lane  = threadIdx.x & 31;
    const int wave  = threadIdx.x >> 5;
    const int mlane = lane & 15;   // A: row within tile, B/C: N within tile
    const int khalf = lane >> 4;   // 0 -> K=0,1 ; 1 -> K=2,3 (per ISA layout)
    const int row_base = blockIdx.x * 16;

    // Preload A fragments: a[t] covers K-chunk [4t, 4t+4)
    v2f a[64];
    const float* zrow = z + (size_t)(row_base + mlane) * DIM + 2 * khalf;
#pragma unroll
    for (int t = 0; t < 64; ++t) a[t] = *(const v2f*)(zrow + 4 * t);

    float bestv[8];
    int   besti[8];
#pragma unroll
    for (int j = 0; j < 8; ++j) { bestv[j] = 3.402823e38f; besti[j] = 0; }

    for (int cb = wave * 16; cb < K_CODES; cb += 8 * 16) {
        v8f acc0 = {}, acc1 = {}, acc2 = {}, acc3 = {};
        const float* erow = emb + (size_t)(cb + mlane) * DIM + 2 * khalf;
#pragma unroll
        for (int t = 0; t < 16; ++t) {
            v2f b0 = *(const v2f*)(erow + 16 * t + 0);
            v2f b1 = *(const v2f*)(erow + 16 * t + 4);
            v2f b2 = *(const v2f*)(erow + 16 * t + 8);
            v2f b3 = *(const v2f*)(erow + 16 * t + 12);
            acc0 = __builtin_amdgcn_wmma_f32_16x16x4_f32(
                false, a[4 * t + 0], false, b0, (short)0, acc0, false, false);
            acc1 = __builtin_amdgcn_wmma_f32_16x16x4_f32(
                false, a[4 * t + 1], false, b1, (short)0, acc1, false, false);
            acc2 = __builtin_amdgcn_wmma_f32_16x16x4_f32(
                false, a[4 * t + 2], false, b2, (short)0, acc2, false, false);
            acc3 = __builtin_amdgcn_wmma_f32_16x16x4_f32(
                false, a[4 * t + 3], false, b3, (short)0, acc3, false, false);
        }
        const float en = enorm[cb + mlane];
        const int   ci = cb + mlane;
#pragma unroll
        for (int j = 0; j < 8; ++j) {
            float dot = (acc0[j] + acc1[j]) + (acc2[j] + acc3[j]);
            float sc = en - 2.0f * dot;
            if (sc < bestv[j]) { bestv[j] = sc; besti[j] = ci; }
        }
    }

    // reduce over the 16 N-lanes (xor within half-wave groups of 16)
#pragma unroll
    for (int j = 0; j < 8; ++j) {
        float v = bestv[j]; int idx = besti[j];
#pragma unroll
        for (int off = 8; off >= 1; off >>= 1) {
            float ov = __shfl_xor(v, off, 32);
            int   oi = __shfl_xor(idx, off, 32);
            if (ov < v || (ov == v && oi < idx)) { v = ov; idx = oi; }
        }
        if (mlane == 0) {
            int row = j + 8 * khalf;   // C layout: lanes 0-15 -> M=j, 16-31 -> M=j+8
            s_val[wave][row] = v;
            s_idx[wave][row] = idx;
        }
    }
    __syncthreads();

    // reduce across the 8 waves
    if (threadIdx.x < 16) {
        float v = s_val[0][threadIdx.x]; int idx = s_idx[0][threadIdx.x];
#pragma unroll
        for (int w = 1; w < 8; ++w) {
            float ov = s_val[w][threadIdx.x]; int oi = s_idx[w][threadIdx.x];
            if (ov < v || (ov == v && oi < idx)) { v = ov; idx = oi; }
        }
        codes_i[row_base + threadIdx.x] = idx;
        codes_f[row_base + threadIdx.x] = (float)idx;
    }
}

// ---------------- gather z_q, loss partials, scatter dw/counts ----------------
__global__ __launch_bounds__(256) void vq_gather(const float* __restrict__ z,
                                                 const float* __restrict__ emb,
                                                 const int* __restrict__ codes,
                                                 float* __restrict__ z_st,
                                                 float* __restrict__ dw,
                                                 float* __restrict__ counts,
                                                 float* __restrict__ loss_acc) {
    __shared__ float sred[256];
    const int n = blockIdx.x;
    const int d = threadIdx.x;
    const int c = codes[n];
    const float zq = emb[(size_t)c * DIM + d];
    const float zv = z[(size_t)n * DIM + d];
    z_st[(size_t)n * DIM + d] = zv + (zq - zv);   // straight-through (matches ref rounding)
    const float diff = zq - zv;
    sred[d] = diff * diff;
    atomicAdd(&dw[(size_t)c * DIM + d], zv);
    if (d == 0) atomicAdd(&counts[c], 1.0f);
    __syncthreads();
    for (int s = 128; s >= 1; s >>= 1) {
        if (d < s) sred[d] += sred[d + s];
        __syncthreads();
    }
    if (d == 0) atomicAdd(loss_acc, sred[0]);
}

// ---------------- EMA cluster size + total ----------------
__global__ __launch_bounds__(1024) void vq_cluster(const float* __restrict__ ema_cs,
                                                   const float* __restrict__ counts,
                                                   float* __restrict__ new_cs,
                                                   float* __restrict__ n_out) {
    __shared__ float sred[1024];
    float s = 0.f;
    for (int k = threadIdx.x; k < K_CODES; k += 1024) {
        float v = ema_cs[k] * DECAY_F + OMD_F * counts[k];
        new_cs[k] = v;
        s += v;
    }
    sred[threadIdx.x] = s;
    __syncthreads();
    for (int st = 512; st >= 1; st >>= 1) {
        if (threadIdx.x < st) sred[threadIdx.x] += sred[threadIdx.x + st];
        __syncthreads();
    }
    if (threadIdx.x == 0) n_out[0] = sred[0];
}

// ---------------- new_ema_w and normalized embedding ----------------
__global__ void vq_finalize(const float* __restrict__ ema_w,
                            const float* __restrict__ dw,
                            const float* __restrict__ new_cs,
                            const float* __restrict__ n_ptr,
                            float* __restrict__ new_emb,
                            float* __restrict__ new_ema_w) {
    int i = blockIdx.x * blockDim.x + threadIdx.x;
    if (i >= K_CODES * DIM) return;
    int k = i >> 8;   // DIM == 256
    float n = n_ptr[0];
    float cs = (new_cs[k] + EPS_F) / (n + (float)K_CODES * EPS_F) * n;
    float nw = ema_w[i] * DECAY_F + OMD_F * dw[i];
    new_ema_w[i] = nw;
    new_emb[i] = nw / cs;
}

__global__ void vq_loss_final(const float* __restrict__ loss_acc, float* __restrict__ out) {
    out[0] = COMMIT_F * loss_acc[0] / (float)((size_t)N_VECS * DIM);
}

extern "C" void kernel_launch(void* const* d_in, const int* in_sizes, int n_in,
                              void* d_out, int out_size, void* d_ws, size_t ws_size,
                              hipStream_t stream) {
    const float* z      = (const float*)d_in[0];
    const float* emb    = (const float*)d_in[1];
    const float* ema_cs = (const float*)d_in[2];
    const float* ema_w  = (const float*)d_in[3];

    // d_out layout: z_st | vq_loss | codes(f32) | new_embedding | new_cluster_size | new_ema_w
    float* out     = (float*)d_out;
    float* o_zst   = out;
    float* o_loss  = o_zst + (size_t)N_VECS * DIM;
    float* o_codes = o_loss + 1;
    float* o_nemb  = o_codes + N_VECS;
    float* o_ncs   = o_nemb + (size_t)K_CODES * DIM;
    float* o_nw    = o_ncs + K_CODES;

    // workspace layout (dw, counts, loss contiguous so one zero pass covers them)
    float* ws       = (float*)d_ws;
    float* w_dw     = ws;                               // K*D
    float* w_counts = w_dw + (size_t)K_CODES * DIM;     // K
    float* w_loss   = w_counts + K_CODES;               // 1
    float* w_n      = w_loss + 1;                       // 1
    float* w_enorm  = w_n + 1;                          // K
    int*   w_codes  = (int*)(w_enorm + K_CODES);        // N ints

    const int nzero = K_CODES * DIM + K_CODES + 1;
    vq_zero<<<(nzero + 255) / 256, 256, 0, stream>>>(w_dw, nzero);
    vq_enorm<<<K_CODES, 32, 0, stream>>>(emb, w_enorm);
    vq_argmin<<<N_VECS / 16, 256, 0, stream>>>(z, emb, w_enorm, w_codes, o_codes);
    vq_gather<<<N_VECS, 256, 0, stream>>>(z, emb, w_codes, o_zst, w_dw, w_counts, w_loss);
    vq_cluster<<<1, 1024, 0, stream>>>(ema_cs, w_counts, o_ncs, w_n);
    vq_finalize<<<(K_CODES * DIM + 255) / 256, 256, 0, stream>>>(ema_w, w_dw, o_ncs, w_n, o_nemb, o_nw);
    vq_loss_final<<<1, 1, 0, stream>>>(w_loss, o_loss);
}